// PatchEncoder_7189775253534
// MI455X (gfx1250) — compile-verified
//
#include <hip/hip_runtime.h>
#include <hip/hip_bf16.h>

typedef float v2f __attribute__((ext_vector_type(2)));
typedef float v8f __attribute__((ext_vector_type(8)));

#define B_DIM    64
#define N_DIM    10000
#define C0_DIM   128
#define C2_DIM   256
#define M_TILES  ((B_DIM * N_DIM) / 16)  // 40000
#define LDS_ROW  132                     // 128 + 4 pad -> conflict-free ds_load_b64
#define LDS_TILE (16 * LDS_ROW)
#define GRID     2048

__device__ __forceinline__ v8f splat8(float x) {
    v8f r;
#pragma unroll
    for (int i = 0; i < 8; ++i) r[i] = x;
    return r;
}

// ---------------------------------------------------------------------------
// Kernel 1: c[j] = be[j] + sum_o k0[o] * We[128+o, j],  k0 = MLP(0)
// (x = coords - coords = 0, so h1 = relu(b1); W1/coordinates are dead)
// ---------------------------------------------------------------------------
__global__ void pe_const_kernel(const float* __restrict__ b1,
                                const float* __restrict__ W2,
                                const float* __restrict__ b2,
                                const float* __restrict__ W3,
                                const float* __restrict__ b3,
                                const float* __restrict__ We,
                                const float* __restrict__ be,
                                float* __restrict__ cvec) {
    const int col = threadIdx.x;  // 256 threads, one output column each
    float h1[8];
#pragma unroll
    for (int i = 0; i < 8; ++i) h1[i] = fmaxf(b1[i], 0.0f);

    float h2[16];
#pragma unroll
    for (int j = 0; j < 16; ++j) {
        float s = b2[j];
#pragma unroll
        for (int i = 0; i < 8; ++i) s = fmaf(h1[i], W2[i * 16 + j], s);
        h2[j] = fmaxf(s, 0.0f);
    }

    float acc = be[col];
    for (int o = 0; o < 64; ++o) {
        float k = b3[o];
#pragma unroll
        for (int j = 0; j < 16; ++j) k = fmaf(h2[j], W3[j * 64 + o], k);
        acc = fmaf(k, We[(C0_DIM + o) * C2_DIM + col], acc);
    }
    cvec[col] = acc;
}

// ---------------------------------------------------------------------------
// Kernel 2: persistent WMMA GEMM, 512 threads = 16 waves.
//   Wave w owns 16 output columns [16w, 16w+16): its B slice (128x16 of We)
//   is 32 x v2f WMMA-B operands = 64 VGPRs (no spill).
//   A tile (16x128) is double-buffered in LDS; one barrier per tile.
// ---------------------------------------------------------------------------
__global__ void __launch_bounds__(512)
pe_gemm_kernel(const float* __restrict__ feature,
               const float* __restrict__ We,
               const float* __restrict__ cvec,
               float* __restrict__ out) {
    __shared__ float lds[2 * LDS_TILE];

    const int tid   = threadIdx.x;
    const int lane  = tid & 31;
    const int wave  = tid >> 5;             // 0..15
    const int l15   = lane & 15;
    const int khalf = (lane < 16) ? 0 : 2;  // K rows held by this lane half
    const int mhalf = (lane < 16) ? 0 : 8;  // M offset in C/D layout
    const int n0    = wave * 16;            // this wave's column block

    // --- register-resident B operands (VGPR0 = K0/K2, VGPR1 = K1/K3 per ISA)
    v2f breg[32];
#pragma unroll
    for (int kk = 0; kk < 32; ++kk) {
        const int kb = 4 * kk + khalf;
        v2f b;
        b.x = We[kb * C2_DIM + n0 + l15];
        b.y = We[(kb + 1) * C2_DIM + n0 + l15];
        breg[kk] = b;
    }
    const float cv = cvec[n0 + l15];

    // staging role: wave w stages row w (contiguous 512B per wave, b128 ops)
    const int srow = tid >> 5;        // 0..15 : tile row
    const int skq  = (tid & 31) * 4;  // 0..124: K offset (float4 per thread)

    // --- preload first tile into buffer 0
    int tile = blockIdx.x;
    {
        const int r0 = tile * 16;
        const float4 p = *(const float4*)(feature +
            ((size_t)((r0 & 63) + srow) * N_DIM + (r0 >> 6)) * C0_DIM + skq);
        *(float4*)&lds[srow * LDS_ROW + skq] = p;
    }

    int buf = 0;
    for (; tile < M_TILES; tile += GRID) {
        __syncthreads();  // staged data for `tile` visible; prev reads done

        // --- issue next tile's global loads early (latency hidden by WMMAs)
        const int  tnext    = tile + GRID;
        const bool has_next = tnext < M_TILES;
        float4 p;
        if (has_next) {
            const int r0n = tnext * 16;
            p = *(const float4*)(feature +
                ((size_t)((r0n & 63) + srow) * N_DIM + (r0n >> 6)) * C0_DIM + skq);
        }

        // --- 32 x v_wmma_f32_16x16x4_f32 from LDS buffer `buf`
        v8f acc = splat8(cv);
        const float* arow = &lds[buf * LDS_TILE + l15 * LDS_ROW + khalf];
#pragma unroll
        for (int kk = 0; kk < 32; ++kk) {
            const v2f a = *(const v2f*)(arow + 4 * kk);  // conflict-free b64
            acc = __builtin_amdgcn_wmma_f32_16x16x4_f32(
                false, a, false, breg[kk], (short)0, acc, false, false);
        }

        // --- stash prefetched tile into the other buffer (not read until
        //     after next barrier)
        if (has_next)
            *(float4*)&lds[(buf ^ 1) * LDS_TILE + srow * LDS_ROW + skq] = p;
        buf ^= 1;

        // --- store: VGPR v -> rows (v, v+8), cols n0 + l15 (2x64B segments)
        const int r0 = tile * 16;
        float* orow = out + (size_t)(r0 + mhalf) * C2_DIM + n0 + l15;
#pragma unroll
        for (int v = 0; v < 8; ++v)
            orow[(size_t)v * C2_DIM] = acc[v];
    }
}

// ---------------------------------------------------------------------------
extern "C" void kernel_launch(void* const* d_in, const int* in_sizes, int n_in,
                              void* d_out, int out_size, void* d_ws, size_t ws_size,
                              hipStream_t stream) {
    // setup_inputs order:
    // 0 feature [B,N,C0], 1 coordinates (unused), 2 W1 (unused), 3 b1,
    // 4 W2, 5 b2, 6 W3, 7 b3, 8 We [192,256], 9 be [256]
    const float* feature = (const float*)d_in[0];
    const float* b1      = (const float*)d_in[3];
    const float* W2      = (const float*)d_in[4];
    const float* b2      = (const float*)d_in[5];
    const float* W3      = (const float*)d_in[6];
    const float* b3      = (const float*)d_in[7];
    const float* We      = (const float*)d_in[8];
    const float* be      = (const float*)d_in[9];
    float*       outp    = (float*)d_out;
    float*       cvec    = (float*)d_ws;  // 256 floats of scratch

    pe_const_kernel<<<1, 256, 0, stream>>>(b1, W2, b2, W3, b3, We, be, cvec);
    pe_gemm_kernel<<<GRID, 512, 0, stream>>>(feature, We, cvec, outp);
}